// GATLayer_90486370992270
// MI455X (gfx1250) — compile-verified
//
#include <hip/hip_runtime.h>
#include <hip/hip_bf16.h>

// ---------------------------------------------------------------------------
// GAT layer for MI455X (gfx1250, wave32).
// K1: feat = x @ fc_w^T via V_WMMA_F32_16X16X4_F32 (fp32-exact matrix core)
// K2: per-(node,head) attention dots el/er; init emax=-inf, denom=0,
//     out = x + bias (residual/bias epilogue folded into the accumulator init)
// K3: edge segment-max (int-punned float atomic max)
// K4: edge segment-sum of exp(e - emax)
// K5: weighted message scatter-add (128 lanes per edge, L2-resident atomics)
// ---------------------------------------------------------------------------

typedef __attribute__((ext_vector_type(2))) float v2f;
typedef __attribute__((ext_vector_type(8))) float v8f;

#define IN_FEATS 128
#define HEADS 4
#define OUT_FEATS 32
#define NEG_SLOPE 0.2f

// ---------------- K1: fp32 WMMA GEMM: feat[n,o] = sum_k x[n,k]*w[o,k] -------
// grid.x = N/16 (row tiles), block = 256 (8 waves); wave w owns col tile w.
__global__ __launch_bounds__(256) void k1_gemm_wmma(
    const float* __restrict__ x, const float* __restrict__ w,
    float* __restrict__ feat) {
  const int lane = threadIdx.x & 31;
  const int wave = threadIdx.x >> 5;
  const int half = lane >> 4;   // 0: K even pair base, 1: K odd pair base
  const int l15  = lane & 15;
  const int row0 = blockIdx.x * 16;
  const int col0 = wave * 16;

  const float* __restrict__ xrow = x + (size_t)(row0 + l15) * IN_FEATS;
  const float* __restrict__ wrow = w + (size_t)(col0 + l15) * IN_FEATS;

  v8f acc = {};
#pragma unroll
  for (int k = 0; k < IN_FEATS; k += 4) {
    // A 16x4 fp32: VGPR0 = {K=k (lanes0-15), K=k+2 (lanes16-31)}, VGPR1 = +1
    v2f a = *(const v2f*)(xrow + k + 2 * half);
    // B 4x16 fp32: same striping, N = l15 across lanes
    v2f b = *(const v2f*)(wrow + k + 2 * half);
    acc = __builtin_amdgcn_wmma_f32_16x16x4_f32(
        /*neg_a=*/false, a, /*neg_b=*/false, b,
        /*c_mod=*/(short)0, acc, /*reuse_a=*/false, /*reuse_b=*/false);
  }

  // D layout: acc[r] -> (M = r + 8*half, N = l15)
#pragma unroll
  for (int r = 0; r < 8; ++r) {
    feat[(size_t)(row0 + r + 8 * half) * IN_FEATS + col0 + l15] = acc[r];
  }
}

// ---------------- K2: node prep -------------------------------------------
__global__ __launch_bounds__(256) void k2_node_prep(
    const float* __restrict__ feat, const float* __restrict__ attn_l,
    const float* __restrict__ attn_r, const float* __restrict__ x,
    const float* __restrict__ bias, float* __restrict__ el,
    float* __restrict__ er, float* __restrict__ emax,
    float* __restrict__ denom, float* __restrict__ out, int N) {
  int t = blockIdx.x * blockDim.x + threadIdx.x;
  if (t >= N * HEADS) return;
  int n = t >> 2, h = t & 3;
  const float* f  = feat + (size_t)n * IN_FEATS + h * OUT_FEATS;
  const float* al = attn_l + h * OUT_FEATS;
  const float* ar = attn_r + h * OUT_FEATS;
  float sl = 0.f, sr = 0.f;
#pragma unroll
  for (int j = 0; j < OUT_FEATS; ++j) {
    float v = f[j];
    sl = fmaf(v, al[j], sl);
    sr = fmaf(v, ar[j], sr);
  }
  el[t] = sl;
  er[t] = sr;
  emax[t]  = __int_as_float(0xFF800000);  // -inf
  denom[t] = 0.f;
  // out accumulator starts at residual + bias
  const float* xr = x + (size_t)n * IN_FEATS + h * OUT_FEATS;
  const float* br = bias + h * OUT_FEATS;
  float* o = out + (size_t)n * IN_FEATS + h * OUT_FEATS;
#pragma unroll
  for (int j = 0; j < OUT_FEATS; ++j) o[j] = xr[j] + br[j];
}

// exact float atomic max via integer ordering trick (init must be -inf bits)
__device__ inline void atomic_max_f32(float* addr, float val) {
  if (!(val < 0.f)) {
    atomicMax((int*)addr, __float_as_int(val));
  } else {
    atomicMin((unsigned int*)addr, (unsigned int)__float_as_int(val));
  }
}

__device__ inline float leaky(float v) { return v > 0.f ? v : NEG_SLOPE * v; }

// ---------------- K3: per-edge segment max --------------------------------
__global__ __launch_bounds__(256) void k3_edge_max(
    const int* __restrict__ src, const int* __restrict__ dst,
    const float* __restrict__ el, const float* __restrict__ er,
    float* __restrict__ emax, int E) {
  int e = blockIdx.x * blockDim.x + threadIdx.x;
  if (e >= E) return;
  int s = src[e], d = dst[e];
  const float4 L = *(const float4*)(el + (size_t)s * HEADS);
  const float4 R = *(const float4*)(er + (size_t)d * HEADS);
  atomic_max_f32(&emax[(size_t)d * HEADS + 0], leaky(L.x + R.x));
  atomic_max_f32(&emax[(size_t)d * HEADS + 1], leaky(L.y + R.y));
  atomic_max_f32(&emax[(size_t)d * HEADS + 2], leaky(L.z + R.z));
  atomic_max_f32(&emax[(size_t)d * HEADS + 3], leaky(L.w + R.w));
}

// ---------------- K4: per-edge exp + segment sum --------------------------
__global__ __launch_bounds__(256) void k4_edge_denom(
    const int* __restrict__ src, const int* __restrict__ dst,
    const float* __restrict__ el, const float* __restrict__ er,
    const float* __restrict__ emax, float* __restrict__ denom, int E) {
  int e = blockIdx.x * blockDim.x + threadIdx.x;
  if (e >= E) return;
  int s = src[e], d = dst[e];
  const float4 L = *(const float4*)(el + (size_t)s * HEADS);
  const float4 R = *(const float4*)(er + (size_t)d * HEADS);
  const float4 M = *(const float4*)(emax + (size_t)d * HEADS);
  atomicAdd(&denom[(size_t)d * HEADS + 0], expf(leaky(L.x + R.x) - M.x));
  atomicAdd(&denom[(size_t)d * HEADS + 1], expf(leaky(L.y + R.y) - M.y));
  atomicAdd(&denom[(size_t)d * HEADS + 2], expf(leaky(L.z + R.z) - M.z));
  atomicAdd(&denom[(size_t)d * HEADS + 3], expf(leaky(L.w + R.w) - M.w));
}

// ---------------- K5: weighted message scatter ----------------------------
// 128 threads per edge (one per output column); block 256 = 2 edges.
__global__ __launch_bounds__(256) void k5_edge_scatter(
    const int* __restrict__ src, const int* __restrict__ dst,
    const float* __restrict__ feat, const float* __restrict__ el,
    const float* __restrict__ er, const float* __restrict__ emax,
    const float* __restrict__ denom, float* __restrict__ out, int E) {
  int e = blockIdx.x * 2 + (threadIdx.x >> 7);
  if (e >= E) return;
  int c = threadIdx.x & 127;        // output column 0..127
  int h = c >> 5;                   // head
  int s = src[e], d = dst[e];
  // wave-uniform scalar part (all 32 lanes share edge+head)
  float v = leaky(el[(size_t)s * HEADS + h] + er[(size_t)d * HEADS + h]);
  float ee = expf(v - emax[(size_t)d * HEADS + h]);
  float a = ee / fmaxf(denom[(size_t)d * HEADS + h], 1e-38f);
  atomicAdd(&out[(size_t)d * IN_FEATS + c],
            feat[(size_t)s * IN_FEATS + c] * a);
}

extern "C" void kernel_launch(void* const* d_in, const int* in_sizes, int n_in,
                              void* d_out, int out_size, void* d_ws,
                              size_t ws_size, hipStream_t stream) {
  const float* x      = (const float*)d_in[0];
  const float* fc_w   = (const float*)d_in[1];
  const float* attn_l = (const float*)d_in[2];
  const float* attn_r = (const float*)d_in[3];
  const float* bias   = (const float*)d_in[4];
  const int*   src    = (const int*)d_in[5];
  const int*   dst    = (const int*)d_in[6];

  const int N = in_sizes[0] / IN_FEATS;   // 100000
  const int E = in_sizes[5];              // 1600000
  float* out = (float*)d_out;

  // workspace carve-up (all fp32): feat | el | er | emax | denom
  float* feat  = (float*)d_ws;
  float* el    = feat + (size_t)N * IN_FEATS;
  float* er    = el + (size_t)N * HEADS;
  float* emax  = er + (size_t)N * HEADS;
  float* denom = emax + (size_t)N * HEADS;

  // K1: WMMA GEMM (N divisible by 16 -> exact tiling)
  k1_gemm_wmma<<<N / 16, 256, 0, stream>>>(x, fc_w, feat);

  // K2: attention dots + init
  int t2 = N * HEADS;
  k2_node_prep<<<(t2 + 255) / 256, 256, 0, stream>>>(
      feat, attn_l, attn_r, x, bias, el, er, emax, denom, out, N);

  // K3/K4: edge softmax statistics
  k3_edge_max<<<(E + 255) / 256, 256, 0, stream>>>(src, dst, el, er, emax, E);
  k4_edge_denom<<<(E + 255) / 256, 256, 0, stream>>>(src, dst, el, er, emax,
                                                     denom, E);

  // K5: weighted scatter-add into out (already holds x + bias)
  k5_edge_scatter<<<(E + 1) / 2, 256, 0, stream>>>(src, dst, feat, el, er,
                                                   emax, denom, out, E);
}